// VAE_51213190037704
// MI455X (gfx1250) — compile-verified
//
#include <hip/hip_runtime.h>

// Problem constants (from reference)
#define NN   100000u   // nodes (divisible by 16)
#define NE   1600000u  // edges
#define DIN  7
#define H1   64
#define H2   32
#define NG   64        // graphs

typedef __attribute__((ext_vector_type(2))) float v2f;
typedef __attribute__((ext_vector_type(8))) float v8f;

// fp32 WMMA: D(16x16) = A(16x4) * B(4x16) + C, exact fp32 math.
__device__ __forceinline__ v8f wmma_f32_k4(v2f a, v2f b, v8f c) {
    return __builtin_amdgcn_wmma_f32_16x16x4_f32(
        /*neg_a=*/false, a, /*neg_b=*/false, b,
        /*c_mod=*/(short)0, c, /*reuse_a=*/false, /*reuse_b=*/false);
}

// ---------------- utility ----------------
__global__ void zero_kernel(float* __restrict__ p, unsigned n) {
    unsigned i = blockIdx.x * blockDim.x + threadIdx.x;
    if (i < n) p[i] = 0.0f;
}

// deg[dst] += 1 over edges (deg buffer pre-zeroed)
__global__ void degree_kernel(const int* __restrict__ dst, float* __restrict__ deg) {
    unsigned e = blockIdx.x * blockDim.x + threadIdx.x;
    if (e < NE) atomicAdd(&deg[dst[e]], 1.0f);
}

// dis[i] = rsqrt(deg[i] + 1)   (in place)
__global__ void dis_kernel(float* __restrict__ deg_dis) {
    unsigned i = blockIdx.x * blockDim.x + threadIdx.x;
    if (i < NN) deg_dis[i] = rsqrtf(deg_dis[i] + 1.0f);
}

// normE[e] = dis[src]*dis[dst]  (precomputed once: saves ~800MB of redundant dis reads)
__global__ void norm_kernel(const int* __restrict__ src, const int* __restrict__ dst,
                            const float* __restrict__ dis, float* __restrict__ normE) {
    unsigned e = blockIdx.x * blockDim.x + threadIdx.x;
    if (e < NE) normE[e] = dis[src[e]] * dis[dst[e]];
}

// pad x (N x 7) -> xpad (N x 8, col 7 = 0): branch-free aligned float2 A-loads in GEMM1
__global__ void pad_x_kernel(const float* __restrict__ x, float* __restrict__ xpad) {
    unsigned i = blockIdx.x * blockDim.x + threadIdx.x;
    if (i >= NN * 8u) return;
    unsigned r = i >> 3, k = i & 7u;
    xpad[i] = (k < DIN) ? x[r * DIN + k] : 0.0f;
}

// pad W1 (7 x 64) -> w1pad (8 x 64, row 7 = 0)
__global__ void pad_w1_kernel(const float* __restrict__ W1, float* __restrict__ w1pad) {
    unsigned i = blockIdx.x * blockDim.x + threadIdx.x;
    if (i >= 8u * H1) return;
    unsigned k = i >> 6, c = i & 63u;
    w1pad[i] = (k < DIN) ? W1[k * H1 + c] : 0.0f;
}

// ---------------- GEMM1: xw1 = xpad @ w1pad  (N x 8) @ (8 x 64), fp32 WMMA
// W1 staged in LDS (2KB); one wave per 16-row M-tile, 8 waves per block.
__global__ void gemm1_kernel(const float* __restrict__ xpad, const float* __restrict__ w1pad,
                             float* __restrict__ xw1) {
    __shared__ float sB[8 * H1];
    for (unsigned t = threadIdx.x; t < 8u * H1; t += blockDim.x) sB[t] = w1pad[t];
    __syncthreads();

    unsigned lane  = threadIdx.x & 31u;
    unsigned mtile = blockIdx.x * (blockDim.x >> 5) + (threadIdx.x >> 5);
    if (mtile >= NN / 16u) return;
    unsigned m0    = mtile * 16u;
    unsigned rlo   = lane & 15u;
    unsigned khalf = (lane >> 4) << 1;   // lanes 0-15 -> K{0,1}; lanes 16-31 -> K{2,3}
    unsigned row   = m0 + rlo;

    v8f acc[4] = {};                     // 16x64 output = 4 N-tiles
#pragma unroll
    for (int ks = 0; ks < 2; ks++) {     // K = 8 (padded)
        unsigned kb = ks * 4 + khalf;    // even -> 8B-aligned
        v2f a = *(const v2f*)(xpad + row * 8u + kb);
#pragma unroll
        for (int nt = 0; nt < 4; nt++) {
            unsigned col = nt * 16 + rlo;
            v2f b;
            b.x = sB[kb * H1 + col];
            b.y = sB[(kb + 1) * H1 + col];
            acc[nt] = wmma_f32_k4(a, b, acc[nt]);
        }
    }
    // C/D layout: VGPR g -> row m0+g (lanes 0-15) / m0+8+g (lanes 16-31)
    unsigned rbase = m0 + ((lane >> 4) << 3);
#pragma unroll
    for (int nt = 0; nt < 4; nt++) {
        unsigned col = nt * 16 + rlo;
#pragma unroll
        for (int g = 0; g < 8; g++)
            xw1[(rbase + g) * H1 + col] = acc[nt][g];
    }
}

// ---------------- scatter layer 1: agg1[dst,h] += xw1[src,h] * normE[e]
// One wave per edge (e = idx>>5 is wave-uniform); each lane gathers a float2
// of the 64-wide source row -> 256B fully-coalesced gather per wave.
__global__ void scatter1_kernel(const int* __restrict__ src, const int* __restrict__ dst,
                                const float* __restrict__ normE,
                                const float* __restrict__ xw1, float* __restrict__ agg1) {
    unsigned idx = blockIdx.x * blockDim.x + threadIdx.x;
    if (idx >= NE * 32u) return;
    unsigned e = idx >> 5;
    unsigned h = (idx & 31u) << 1;
    unsigned s = (unsigned)src[e], d = (unsigned)dst[e];
    float nrm = normE[e];
    v2f v = *(const v2f*)(xw1 + s * (unsigned)H1 + h);
    atomicAdd(&agg1[d * (unsigned)H1 + h],     v.x * nrm);
    atomicAdd(&agg1[d * (unsigned)H1 + h + 1], v.y * nrm);
}

// hidden1 = relu(agg + xw1*dis^2 + b1)   (in place over agg1)
__global__ void hidden_fin_kernel(const float* __restrict__ xw1, const float* __restrict__ dis,
                                  const float* __restrict__ b1, float* __restrict__ h1) {
    unsigned idx = blockIdx.x * blockDim.x + threadIdx.x;
    if (idx >= NN * (unsigned)H1) return;
    unsigned i = idx >> 6, h = idx & 63u;
    float ds = dis[i];
    float v  = h1[idx] + xw1[idx] * ds * ds + b1[h];
    h1[idx]  = fmaxf(v, 0.0f);
}

// ---------------- GEMM2+3 fused: hw2 = h1@W2, hw3 = h1@W3  (N x 64)@(64 x 32)
// W2/W3 staged in LDS (16KB); A loads shared between both products.
__global__ void gemm23_kernel(const float* __restrict__ h1,
                              const float* __restrict__ W2, const float* __restrict__ W3,
                              float* __restrict__ hw2, float* __restrict__ hw3) {
    __shared__ float sW2[H1 * H2];
    __shared__ float sW3[H1 * H2];
    for (unsigned t = threadIdx.x; t < (unsigned)(H1 * H2); t += blockDim.x) {
        sW2[t] = W2[t];
        sW3[t] = W3[t];
    }
    __syncthreads();

    unsigned lane  = threadIdx.x & 31u;
    unsigned mtile = blockIdx.x * (blockDim.x >> 5) + (threadIdx.x >> 5);
    if (mtile >= NN / 16u) return;
    unsigned m0    = mtile * 16u;
    unsigned rlo   = lane & 15u;
    unsigned khalf = (lane >> 4) << 1;
    unsigned row   = m0 + rlo;

    v8f acc2[2] = {}, acc3[2] = {};      // two 16-col tiles each (H2 = 32)
#pragma unroll
    for (int ks = 0; ks < 16; ks++) {    // K = 64
        unsigned kb = ks * 4 + khalf;    // even -> 8B-aligned
        v2f a = *(const v2f*)(h1 + row * (unsigned)H1 + kb);
#pragma unroll
        for (int nt = 0; nt < 2; nt++) {
            unsigned col = nt * 16 + rlo;
            v2f b2; b2.x = sW2[kb * H2 + col]; b2.y = sW2[(kb + 1) * H2 + col];
            acc2[nt] = wmma_f32_k4(a, b2, acc2[nt]);
            v2f b3; b3.x = sW3[kb * H2 + col]; b3.y = sW3[(kb + 1) * H2 + col];
            acc3[nt] = wmma_f32_k4(a, b3, acc3[nt]);
        }
    }
    unsigned rbase = m0 + ((lane >> 4) << 3);
#pragma unroll
    for (int nt = 0; nt < 2; nt++) {
        unsigned col = nt * 16 + rlo;
#pragma unroll
        for (int g = 0; g < 8; g++) {
            hw2[(rbase + g) * H2 + col] = acc2[nt][g];
            hw3[(rbase + g) * H2 + col] = acc3[nt][g];
        }
    }
}

// ---------------- scatter layer 2: mu/logvar aggregation fused
// One wave per edge; lane c handles feature c for BOTH mu and logvar.
__global__ void scatter2_kernel(const int* __restrict__ src, const int* __restrict__ dst,
                                const float* __restrict__ normE,
                                const float* __restrict__ hw2, const float* __restrict__ hw3,
                                float* __restrict__ mu_agg, float* __restrict__ lv_agg) {
    unsigned idx = blockIdx.x * blockDim.x + threadIdx.x;
    if (idx >= NE * 32u) return;
    unsigned e = idx >> 5;
    unsigned c = idx & 31u;
    unsigned s = (unsigned)src[e], d = (unsigned)dst[e];
    float nrm = normE[e];
    atomicAdd(&mu_agg[d * (unsigned)H2 + c], hw2[s * (unsigned)H2 + c] * nrm);
    atomicAdd(&lv_agg[d * (unsigned)H2 + c], hw3[s * (unsigned)H2 + c] * nrm);
}

// ---------------- global mean pool of raw x over graphs
__global__ void pool_scatter_kernel(const float* __restrict__ x, const int* __restrict__ batch,
                                    float* __restrict__ xp, float* __restrict__ counts) {
    unsigned i = blockIdx.x * blockDim.x + threadIdx.x;
    if (i >= NN) return;
    int g = batch[i];
    atomicAdd(&counts[g], 1.0f);
#pragma unroll
    for (int d = 0; d < DIN; d++)
        atomicAdd(&xp[g * DIN + d], x[i * DIN + d]);
}

__global__ void pool_div_kernel(float* __restrict__ xp, const float* __restrict__ counts) {
    unsigned idx = threadIdx.x;
    if (idx >= NG * DIN) return;
    xp[idx] /= fmaxf(counts[idx / DIN], 1.0f);
}

// ---------------- final: mu/logvar relu + self-loop + bias, z = eps*exp(logvar)+mu
__global__ void final_kernel(const float* __restrict__ hw2, const float* __restrict__ hw3,
                             const float* __restrict__ dis,
                             const float* __restrict__ b2, const float* __restrict__ b3,
                             const float* __restrict__ eps,
                             float* __restrict__ z, float* __restrict__ mu, float* __restrict__ lv) {
    unsigned idx = blockIdx.x * blockDim.x + threadIdx.x;
    if (idx >= NN * (unsigned)H2) return;
    unsigned i = idx >> 5, h = idx & 31u;
    float ds = dis[i];
    float d2 = ds * ds;
    float m = fmaxf(mu[idx] + hw2[idx] * d2 + b2[h], 0.0f);
    float l = fmaxf(lv[idx] + hw3[idx] * d2 + b3[h], 0.0f);
    mu[idx] = m;
    lv[idx] = l;
    z[idx]  = eps[idx] * expf(l) + m;
}

extern "C" void kernel_launch(void* const* d_in, const int* in_sizes, int n_in,
                              void* d_out, int out_size, void* d_ws, size_t ws_size,
                              hipStream_t stream) {
    const float* x    = (const float*)d_in[0];
    const int*   ei   = (const int*)  d_in[1];   // (2, E): row0 = src, row1 = dst
    const int*   bat  = (const int*)  d_in[2];
    const float* W1   = (const float*)d_in[3];
    const float* b1   = (const float*)d_in[4];
    const float* W2   = (const float*)d_in[5];
    const float* b2   = (const float*)d_in[6];
    const float* W3   = (const float*)d_in[7];
    const float* b3   = (const float*)d_in[8];
    const float* eps  = (const float*)d_in[9];

    const int* src = ei;
    const int* dst = ei + NE;

    // workspace layout (floats), ~88MB total
    float* ws     = (float*)d_ws;
    float* dis    = ws;                          // N      (deg -> dis in place)
    float* normE  = dis   + NN;                  // E
    float* xpad   = normE + NE;                  // N*8
    float* w1pad  = xpad  + NN * 8u;             // 8*H1
    float* xw1    = w1pad + 8u * H1;             // N*H1
    float* agg1   = xw1   + NN * (unsigned)H1;   // N*H1  (becomes hidden1)
    float* hw2    = agg1  + NN * (unsigned)H1;   // N*H2
    float* hw3    = hw2   + NN * (unsigned)H2;   // N*H2
    float* counts = hw3   + NN * (unsigned)H2;   // G

    // output layout (return order: z, x_pooled, mu, logvar)
    float* out = (float*)d_out;
    float* z   = out;                            // N*H2
    float* xp  = z  + NN * (unsigned)H2;         // G*DIN
    float* mu  = xp + NG * DIN;                  // N*H2  (scatter accumulator)
    float* lv  = mu + NN * (unsigned)H2;         // N*H2  (scatter accumulator)

    const int T = 256;
    auto blocks = [](unsigned n, unsigned t) { return (n + t - 1) / t; };

    // zero accumulators (xp..lv is one contiguous d_out region)
    zero_kernel<<<blocks(NN, T), T, 0, stream>>>(dis, NN);
    zero_kernel<<<blocks(NN * (unsigned)H1, T), T, 0, stream>>>(agg1, NN * (unsigned)H1);
    zero_kernel<<<blocks(NG * DIN + 2u * NN * H2, T), T, 0, stream>>>(xp, NG * DIN + 2u * NN * H2);
    zero_kernel<<<1, T, 0, stream>>>(counts, NG);

    // degrees -> dis -> per-edge norm; input padding
    degree_kernel<<<blocks(NE, T), T, 0, stream>>>(dst, dis);
    dis_kernel<<<blocks(NN, T), T, 0, stream>>>(dis);
    norm_kernel<<<blocks(NE, T), T, 0, stream>>>(src, dst, dis, normE);
    pad_x_kernel<<<blocks(NN * 8u, T), T, 0, stream>>>(x, xpad);
    pad_w1_kernel<<<1, 512, 0, stream>>>(W1, w1pad);

    // layer 1
    const unsigned MT = NN / 16u;                // 6250 M-tiles, 8 waves/block
    gemm1_kernel<<<blocks(MT, 8), T, 0, stream>>>(xpad, w1pad, xw1);
    scatter1_kernel<<<blocks(NE * 32u, T), T, 0, stream>>>(src, dst, normE, xw1, agg1);
    hidden_fin_kernel<<<blocks(NN * (unsigned)H1, T), T, 0, stream>>>(xw1, dis, b1, agg1);

    // layers 2/3 (fused)
    gemm23_kernel<<<blocks(MT, 8), T, 0, stream>>>(agg1, W2, W3, hw2, hw3);
    scatter2_kernel<<<blocks(NE * 32u, T), T, 0, stream>>>(src, dst, normE, hw2, hw3, mu, lv);

    // pooling
    pool_scatter_kernel<<<blocks(NN, T), T, 0, stream>>>(x, bat, xp, counts);
    pool_div_kernel<<<1, 512, 0, stream>>>(xp, counts);

    // finalize mu/logvar/z
    final_kernel<<<blocks(NN * (unsigned)H2, T), T, 0, stream>>>(hw2, hw3, dis, b2, b3, eps, z, mu, lv);
}